// MultiEmbedding_2508260901911
// MI455X (gfx1250) — compile-verified
//
#include <hip/hip_runtime.h>
#include <stdint.h>

// MultiEmbedding on MI455X (gfx1250):
//   y[t,:] = sum_{l=0..7} weight[l, x[t,l], :]   (D=1024, f32)
//
// Memory-bound op: 32MB weight table lives in L2 (192MB), HBM traffic is
// dominated by the 536MB output stream. Strategy:
//   * TDM gather-mode descriptor: one tensor_load_to_lds gathers all 8
//     indexed 4KB weight rows (row index = l*1024 + x[t,l], 16-bit indices,
//     2D tensor view 8192x1024) into a 32KB LDS tile, tracked by TENSORcnt.
//   * Double-buffered (2 x 32KB LDS) so the DMA for row t+1 overlaps the
//     reduce+store of row t.
//   * Reduce from LDS with float4 ds_load_b128, stream result with
//     non-temporal stores so the weight table stays resident in L2.
// WMMA is deliberately not used: the op has 0.25 FLOP/byte, there is no
// matmul to map onto v_wmma, and a ones-matrix trick would only add traffic.

typedef uint32_t v4u __attribute__((ext_vector_type(4)));
typedef uint32_t v8u __attribute__((ext_vector_type(8)));
typedef float    vf4 __attribute__((ext_vector_type(4)));

#define NLVL 8
#define DDIM 1024

__device__ __forceinline__ void tdm_gather_row(const int* __restrict__ xp,
                                               uint64_t wbase,
                                               uint32_t lds_byte_addr)
{
    // Pack 8 strictly-increasing 16-bit row indices: idx_l = l*1024 + x_l (< 8192).
    uint32_t p0 = ((uint32_t)xp[0])          | (((uint32_t)xp[1] + 1024u) << 16);
    uint32_t p1 = ((uint32_t)xp[2] + 2048u)  | (((uint32_t)xp[3] + 3072u) << 16);
    uint32_t p2 = ((uint32_t)xp[4] + 4096u)  | (((uint32_t)xp[5] + 5120u) << 16);
    uint32_t p3 = ((uint32_t)xp[6] + 6144u)  | (((uint32_t)xp[7] + 7168u) << 16);
    // Force SGPR placement (values are wave-uniform: one output row per block).
    p0 = (uint32_t)__builtin_amdgcn_readfirstlane((int)p0);
    p1 = (uint32_t)__builtin_amdgcn_readfirstlane((int)p1);
    p2 = (uint32_t)__builtin_amdgcn_readfirstlane((int)p2);
    p3 = (uint32_t)__builtin_amdgcn_readfirstlane((int)p3);

    // D# group 0 (128b): count=1 | gather_mode=1 (16-bit idx) | lds_addr |
    //                    global_addr[56:0] | type=2 ("image")
    v4u g0 = { 0x80000001u,
               lds_byte_addr,
               (uint32_t)wbase,
               ((uint32_t)(wbase >> 32) & 0x01FFFFFFu) | 0x80000000u };

    // D# group 1 (256b):
    //  dw0: workgroup_mask=0 (not in cluster), data_size=2 (4B)
    //  dw1[31:16] = tensor_dim0[15:0]  = 1024
    //  dw2[31:16] = tensor_dim1[15:0]  = 8192 (8 levels * 1024 rows)
    //  dw3[31:16] = tile_dim0          = 1024 elements per row
    //  dw4[15:0]  = tile_dim1          = 8 valid gather indices
    //  dw5        = tensor_dim0_stride = 1024
    v8u g1 = { 0x00020000u,
               (uint32_t)DDIM << 16,
               (uint32_t)(NLVL * DDIM) << 16,
               (uint32_t)DDIM << 16,
               (uint32_t)NLVL,
               (uint32_t)DDIM,
               0u, 0u };

    v4u g2 = { p0, p1, p2, p3 };   // gather row indices 0..7 (16-bit mode)
    v4u g3 = { 0u, 0u, 0u, 0u };   // rows 8..15 unused (tile_dim1 = 8)

    asm volatile("tensor_load_to_lds %0, %1, %2, %3"
                 :
                 : "s"(g0), "s"(g1), "s"(g2), "s"(g3)
                 : "memory");
}

__global__ void __launch_bounds__(256)
multi_embed_tdm_kernel(const int* __restrict__ x1,
                       const int* __restrict__ x2,
                       const float* __restrict__ w,
                       float* __restrict__ out,
                       int t1, int total, int rows_per_block)
{
    __shared__ float buf[2][NLVL * DDIM];   // 2 x 32KB of the 320KB WGP LDS

    const int tid = (int)threadIdx.x;
    // True scalar-branch wave-0 gate: tensor ops ignore EXEC, so the TDM
    // issue must sit behind a uniform (SGPR) condition, not EXEC masking.
    const bool issuer =
        (__builtin_amdgcn_readfirstlane((int)threadIdx.x) >> 5) == 0;

    const int row_beg = (int)blockIdx.x * rows_per_block;
    int row_end = row_beg + rows_per_block;
    if (row_end > total) row_end = total;
    const int nrows = row_end - row_beg;
    if (nrows <= 0) return;

    const uint64_t wbase = (uint64_t)(uintptr_t)w;
    // Low 32 bits of a generic pointer into LDS == byte offset in LDS space.
    const uint32_t lds0 = (uint32_t)(uintptr_t)&buf[0][0];
    const uint32_t lds1 = (uint32_t)(uintptr_t)&buf[1][0];

    // Prologue: kick off the DMA for the first row into buffer 0.
    if (issuer) {
        const int t = row_beg;
        const int* xp = (t < t1) ? (x1 + (size_t)t * NLVL)
                                 : (x2 + (size_t)(t - t1) * NLVL);
        tdm_gather_row(xp, wbase, lds0);
    }

    for (int i = 0; i < nrows; ++i) {
        const int t = row_beg + i;
        const float* cur = buf[i & 1];

        if (issuer) {
            // Wait for the tile feeding this iteration (sole outstanding op).
            __builtin_amdgcn_s_wait_tensorcnt(0);
        }
        __syncthreads();   // publish LDS tile; also fences last iter's readers

        // Overlap: start DMA for the next row into the other buffer.
        if (issuer && (i + 1 < nrows)) {
            const int tn = t + 1;
            const int* xp = (tn < t1) ? (x1 + (size_t)tn * NLVL)
                                      : (x2 + (size_t)(tn - t1) * NLVL);
            tdm_gather_row(xp, wbase, (i & 1) ? lds0 : lds1);
        }

        // 256 threads x float4 == 1024 columns: 8 x ds_load_b128 + reduce.
        vf4 acc = (vf4)(0.0f);
#pragma unroll
        for (int l = 0; l < NLVL; ++l) {
            vf4 v = *(const vf4*)(cur + l * DDIM + (tid << 2));
            acc += v;
        }
        // Stream the output; keep the 32MB weight table resident in L2.
        __builtin_nontemporal_store(
            acc, (vf4*)(out + (size_t)t * DDIM + (tid << 2)));
    }
}

extern "C" void kernel_launch(void* const* d_in, const int* in_sizes, int n_in,
                              void* d_out, int out_size, void* d_ws, size_t ws_size,
                              hipStream_t stream) {
    (void)n_in; (void)out_size; (void)d_ws; (void)ws_size;
    const int*   x1 = (const int*)d_in[0];
    const int*   x2 = (const int*)d_in[1];
    const float* w  = (const float*)d_in[2];
    float* out = (float*)d_out;

    const int t1    = in_sizes[0] / NLVL;   // 65536
    const int t2    = in_sizes[1] / NLVL;   // 65536
    const int total = t1 + t2;              // 131072 output rows

    const int grid = 4096;                  // 64KB LDS/block -> 5 blocks/WGP
    const int rows_per_block = (total + grid - 1) / grid;

    multi_embed_tdm_kernel<<<grid, 256, 0, stream>>>(
        x1, x2, w, out, t1, total, rows_per_block);
}